// MyGNN_40011915329795
// MI455X (gfx1250) — compile-verified
//
#include <hip/hip_runtime.h>
#include <math.h>

typedef __attribute__((ext_vector_type(2))) float v2f;
typedef __attribute__((ext_vector_type(8))) float v8f;

#define TPB 256

__device__ __forceinline__ float lrelu02(float v){ return v > 0.f ? v : 0.2f*v; }
__device__ __forceinline__ float reluf(float v){ return v > 0.f ? v : 0.f; }

// float atomic max via monotone int mapping (works for all finite floats)
__device__ __forceinline__ void atomicMaxF(float* addr, float val){
  if (val >= 0.f) atomicMax((int*)addr, __float_as_int(val));
  else            atomicMin((unsigned int*)addr, __float_as_uint(val));
}

// ---------------- reduction scratch helpers ----------------
// red[0..3]=sum, red[4..7]=sumsq, red[8..11]=bn scale, red[12..15]=bn shift
__global__ void k_zero8(float* red){ if (threadIdx.x < 8) red[threadIdx.x] = 0.f; }

__global__ void k_bn_fin(float* red, const float* g, const float* be, float invn){
  int c = threadIdx.x;
  if (c < 4){
    float mean = red[c]*invn;
    float var  = red[4+c]*invn - mean*mean;
    float sc   = g[c]*rsqrtf(var + 1e-5f);
    red[8+c]  = sc;
    red[12+c] = be[c] - mean*sc;
    red[c] = 0.f; red[4+c] = 0.f;   // re-arm sums for the next BN pass
  }
}

// ---------------- encoder ----------------
// z = x(16) @ W1(16x4) + b1 ; accumulate per-channel sum/sumsq
__global__ void k_enc1(const float* __restrict__ x, const float* __restrict__ W,
                       const float* __restrict__ b, float* __restrict__ z,
                       float* red, int n){
  __shared__ float ssum[8];
  if (threadIdx.x < 8) ssum[threadIdx.x] = 0.f;
  __syncthreads();
  int i = blockIdx.x*blockDim.x + threadIdx.x;
  if (i < n){
    float zv[4];
#pragma unroll
    for (int c=0;c<4;c++) zv[c] = b[c];
#pragma unroll
    for (int k=0;k<16;k++){
      float xv = x[(long)i*16 + k];
#pragma unroll
      for (int c=0;c<4;c++) zv[c] += xv * W[k*4+c];
    }
#pragma unroll
    for (int c=0;c<4;c++){
      z[(long)i*4+c] = zv[c];
      atomicAdd(&ssum[c],   zv[c]);
      atomicAdd(&ssum[4+c], zv[c]*zv[c]);
    }
  }
  __syncthreads();
  if (threadIdx.x < 8) atomicAdd(&red[threadIdx.x], ssum[threadIdx.x]);
}

// h = relu(bn(z)); z2 = h @ W2(4x4) + b2  (in place); accumulate stats
__global__ void k_enc2(float* __restrict__ z, const float* __restrict__ W,
                       const float* __restrict__ b, float* red, int n){
  __shared__ float ssum[8];
  if (threadIdx.x < 8) ssum[threadIdx.x] = 0.f;
  __syncthreads();
  int i = blockIdx.x*blockDim.x + threadIdx.x;
  if (i < n){
    float h[4];
#pragma unroll
    for (int k=0;k<4;k++) h[k] = reluf(z[(long)i*4+k]*red[8+k] + red[12+k]);
#pragma unroll
    for (int c=0;c<4;c++){
      float zv = b[c];
#pragma unroll
      for (int k=0;k<4;k++) zv += h[k]*W[k*4+c];
      z[(long)i*4+c] = zv;
      atomicAdd(&ssum[c],   zv);
      atomicAdd(&ssum[4+c], zv*zv);
    }
  }
  __syncthreads();
  if (threadIdx.x < 8) atomicAdd(&red[threadIdx.x], ssum[threadIdx.x]);
}

// h2 = relu(bn(z2)); h0 = h2 @ W3(4x12) + b3
__global__ void k_enc3(const float* __restrict__ z, const float* __restrict__ W,
                       const float* __restrict__ b, const float* __restrict__ red,
                       float* __restrict__ h0, int n){
  int i = blockIdx.x*blockDim.x + threadIdx.x;
  if (i >= n) return;
  float h[4];
#pragma unroll
  for (int k=0;k<4;k++) h[k] = reluf(z[(long)i*4+k]*red[8+k] + red[12+k]);
#pragma unroll
  for (int j=0;j<12;j++){
    float v = b[j];
#pragma unroll
    for (int k=0;k<4;k++) v += h[k]*W[k*12+j];
    h0[(long)i*12+j] = v;
  }
}

// ---------------- GATv2 layers ----------------
// xl[n][r*4+c] = h @ Wl[r] + bl[r] ; xr likewise (all 3 relations)
__global__ void k_node_lin(const float* __restrict__ hin, int hstride,
                           const float* __restrict__ Wl, const float* __restrict__ bl,
                           const float* __restrict__ Wr, const float* __restrict__ br,
                           float* __restrict__ xl, float* __restrict__ xr, int n){
  int i = blockIdx.x*blockDim.x + threadIdx.x;
  if (i >= n) return;
  float h[12];
#pragma unroll
  for (int k=0;k<12;k++) h[k] = hin[(long)i*hstride + k];
#pragma unroll
  for (int r=0;r<3;r++){
#pragma unroll
    for (int c=0;c<4;c++){
      float al = bl[r*4+c], ar = br[r*4+c];
#pragma unroll
      for (int k=0;k<12;k++){
        al += h[k]*Wl[(r*12+k)*4+c];
        ar += h[k]*Wr[(r*12+k)*4+c];
      }
      xl[(long)i*12 + r*4 + c] = al;
      xr[(long)i*12 + r*4 + c] = ar;
    }
  }
}

__global__ void k_seg_init(float* smax, float* den, float* acc, int roff, int n){
  int i = blockIdx.x*blockDim.x + threadIdx.x;
  if (i >= n) return;
  smax[i] = -3.402823466e38f;
  den[i]  = 0.f;
#pragma unroll
  for (int c=0;c<4;c++) acc[(long)i*12 + roff + c] = 0.f;
}

// pass 1: score = leaky(xl[s]+xr[d]) . att ; segment max
__global__ void k_edge1(const int* __restrict__ ei, int E,
                        const float* __restrict__ xl, const float* __restrict__ xr,
                        const float* __restrict__ att, float* __restrict__ sc,
                        float* smax, int roff){
  int e = blockIdx.x*blockDim.x + threadIdx.x;
  if (e >= E) return;
  int s = ei[e], d = ei[E+e];
  float score = 0.f;
#pragma unroll
  for (int c=0;c<4;c++){
    float v = lrelu02(xl[(long)s*12 + roff + c] + xr[(long)d*12 + roff + c]);
    score += v*att[c];
  }
  sc[e] = score;
  atomicMaxF(&smax[d], score);
}

// pass 2: a = exp(score - smax[d]) ; segment sum
__global__ void k_edge2(const int* __restrict__ ei, int E, float* __restrict__ sc,
                        const float* __restrict__ smax, float* den){
  int e = blockIdx.x*blockDim.x + threadIdx.x;
  if (e >= E) return;
  int d = ei[E+e];
  float a = expf(sc[e] - smax[d]);
  sc[e] = a;
  atomicAdd(&den[d], a);
}

// pass 3: acc[d] += xl[s] * (a/den[d])
__global__ void k_edge3(const int* __restrict__ ei, int E,
                        const float* __restrict__ xl, const float* __restrict__ sc,
                        const float* __restrict__ den, float* acc, int roff){
  int e = blockIdx.x*blockDim.x + threadIdx.x;
  if (e >= E) return;
  int s = ei[e], d = ei[E+e];
  float attn = sc[e] / den[d];
#pragma unroll
  for (int c=0;c<4;c++)
    atomicAdd(&acc[(long)d*12 + roff + c], xl[(long)s*12 + roff + c]*attn);
}

// h_new (into catx columns) = relu(acc + bias)
__global__ void k_node_out(const float* __restrict__ acc, const float* __restrict__ bias,
                           float* __restrict__ catcol, int n){
  int i = blockIdx.x*blockDim.x + threadIdx.x;
  if (i >= n) return;
#pragma unroll
  for (int j=0;j<12;j++)
    catcol[(long)i*36 + j] = reluf(acc[(long)i*12 + j] + bias[j]);
}

// ---------------- decoder ----------------
// prod = catx[src]*catx[dst] (2M x 36); z = prod @ W1(36x4) + b1 via
// V_WMMA_F32_16X16X4_F32 (16 rows per wave, 9-step K chain). Accumulate BN stats.
__global__ void k_dec1_wmma(const int* __restrict__ ei, int E,
                            const float* __restrict__ catx,
                            const float* __restrict__ W1, const float* __restrict__ b1,
                            float* __restrict__ z, float* red){
  __shared__ float lds[8*16*37];
  __shared__ float ssum[8];
  int tid  = threadIdx.x;
  int wave = tid >> 5, lane = tid & 31;
  if (tid < 8) ssum[tid] = 0.f;
  float* tile = &lds[wave*16*37];
  long base = (long)blockIdx.x*128 + (long)wave*16;

  // stage 16x36 prod tile (row-padded to 37 floats)
  for (int idx = lane; idx < 16*36; idx += 32){
    int m = idx/36, k = idx - m*36;
    long e = base + m; if (e >= E) e = E-1;
    int s = ei[e], d = ei[E+e];
    tile[m*37 + k] = catx[(long)s*36 + k]*catx[(long)d*36 + k];
  }
  __syncthreads();

  // A layout (16x4 f32): lanes 0-15 hold K = k0,k0+1; lanes 16-31 hold K = k0+2,k0+3
  // B layout (4x16):     lanes hold col N = lane&15; K split by lane half, same as A
  int half = (lane < 16) ? 0 : 2;
  int row  = lane & 15;
  int col  = lane & 15;
  int cc   = col & 3;                 // in-bounds weight column
  v8f acc = {0.f,0.f,0.f,0.f,0.f,0.f,0.f,0.f};
#pragma unroll
  for (int step = 0; step < 9; step++){
    int k0 = step*4;
    v2f a, b;
    a.x = tile[row*37 + k0 + half];
    a.y = tile[row*37 + k0 + half + 1];
    float w0 = W1[(k0 + half)*4 + cc];
    float w1 = W1[(k0 + half + 1)*4 + cc];
    b.x = (col < 4) ? w0 : 0.f;
    b.y = (col < 4) ? w1 : 0.f;
    acc = __builtin_amdgcn_wmma_f32_16x16x4_f32(false, a, false, b, (short)0, acc,
                                                false, false);
  }

  // D layout: lane<16 -> rows 0..7 (vgpr v = row), lane>=16 -> rows 8..15; col = lane&15
  if (col < 4){
    int rowbase = (lane < 16) ? 0 : 8;
    float bb = b1[col];
    float lsum = 0.f, lsq = 0.f;
#pragma unroll
    for (int v=0; v<8; v++){
      long e = base + rowbase + v;
      if (e < E){
        float val = acc[v] + bb;
        z[e*4 + col] = val;
        lsum += val; lsq += val*val;
      }
    }
    atomicAdd(&ssum[col],   lsum);
    atomicAdd(&ssum[4+col], lsq);
  }
  __syncthreads();
  if (tid < 8) atomicAdd(&red[tid], ssum[tid]);
}

// h = relu(bn(z)); z2 = h @ W2(4x4) + b2 (in place); accumulate stats
__global__ void k_dec2(float* __restrict__ z, const float* __restrict__ W,
                       const float* __restrict__ b, float* red, int n){
  __shared__ float ssum[8];
  if (threadIdx.x < 8) ssum[threadIdx.x] = 0.f;
  __syncthreads();
  int i = blockIdx.x*blockDim.x + threadIdx.x;
  if (i < n){
    float h[4];
#pragma unroll
    for (int k=0;k<4;k++) h[k] = reluf(z[(long)i*4+k]*red[8+k] + red[12+k]);
#pragma unroll
    for (int c=0;c<4;c++){
      float zv = b[c];
#pragma unroll
      for (int k=0;k<4;k++) zv += h[k]*W[k*4+c];
      z[(long)i*4+c] = zv;
      atomicAdd(&ssum[c],   zv);
      atomicAdd(&ssum[4+c], zv*zv);
    }
  }
  __syncthreads();
  if (threadIdx.x < 8) atomicAdd(&red[threadIdx.x], ssum[threadIdx.x]);
}

// h2 = relu(bn(z2)); o = h2 @ W3(4x1) + b3
__global__ void k_dec3(const float* __restrict__ z, const float* __restrict__ W,
                       const float* __restrict__ b, const float* __restrict__ red,
                       float* __restrict__ o, int n){
  int i = blockIdx.x*blockDim.x + threadIdx.x;
  if (i >= n) return;
  float v = b[0];
#pragma unroll
  for (int k=0;k<4;k++) v += reluf(z[(long)i*4+k]*red[8+k] + red[12+k]) * W[k];
  o[i] = v;
}

// pairwise softmax over consecutive edges
__global__ void k_softmax(const float* __restrict__ o, float* __restrict__ out, int npairs){
  int i = blockIdx.x*blockDim.x + threadIdx.x;
  if (i >= npairs) return;
  float o0 = o[2*i], o1 = o[2*i+1];
  float m = fmaxf(o0, o1);
  float e0 = expf(o0 - m), e1 = expf(o1 - m);
  float inv = 1.f/(e0 + e1);
  out[2*i]   = e0*inv;
  out[2*i+1] = e1*inv;
}

// ---------------- launcher ----------------
extern "C" void kernel_launch(void* const* d_in, const int* in_sizes, int n_in,
                              void* d_out, int out_size, void* d_ws, size_t ws_size,
                              hipStream_t stream) {
  const float* x     = (const float*)d_in[0];
  const int*   ei_p  = (const int*)d_in[1];
  const int*   ei_r  = (const int*)d_in[2];
  const int*   ei_c  = (const int*)d_in[3];
  const float* enc_W1 = (const float*)d_in[4];
  const float* enc_b1 = (const float*)d_in[5];
  const float* enc_g1 = (const float*)d_in[6];
  const float* enc_be1= (const float*)d_in[7];
  const float* enc_W2 = (const float*)d_in[8];
  const float* enc_b2 = (const float*)d_in[9];
  const float* enc_g2 = (const float*)d_in[10];
  const float* enc_be2= (const float*)d_in[11];
  const float* enc_W3 = (const float*)d_in[12];
  const float* enc_b3 = (const float*)d_in[13];
  const float* conv_Wl  = (const float*)d_in[14];
  const float* conv_bl  = (const float*)d_in[15];
  const float* conv_Wr  = (const float*)d_in[16];
  const float* conv_br  = (const float*)d_in[17];
  const float* conv_att = (const float*)d_in[18];
  const float* conv_bias= (const float*)d_in[19];
  const float* dec_W1 = (const float*)d_in[20];
  const float* dec_b1 = (const float*)d_in[21];
  const float* dec_g1 = (const float*)d_in[22];
  const float* dec_be1= (const float*)d_in[23];
  const float* dec_W2 = (const float*)d_in[24];
  const float* dec_b2 = (const float*)d_in[25];
  const float* dec_g2 = (const float*)d_in[26];
  const float* dec_be2= (const float*)d_in[27];
  const float* dec_W3 = (const float*)d_in[28];
  const float* dec_b3 = (const float*)d_in[29];

  const int N  = in_sizes[0] / 16;
  const int EP = in_sizes[1] / 2;
  const int EC = in_sizes[3] / 2;

  // workspace layout (floats)
  float* ws = (float*)d_ws;
  size_t o = 0;
  float* red   = ws + o; o += 32;
  float* encz  = ws + o; o += (size_t)N*4;
  float* h0    = ws + o; o += (size_t)N*12;
  float* catx  = ws + o; o += (size_t)N*36;
  float* xl    = ws + o; o += (size_t)N*12;
  float* xr    = ws + o; o += (size_t)N*12;
  float* smax  = ws + o; o += (size_t)N;
  float* den   = ws + o; o += (size_t)N;
  float* accb  = ws + o; o += (size_t)N*12;
  float* etmp  = ws + o; o += (size_t)EC;      // per-edge score/a, reused for dec logits
  float* decz  = ws + o; o += (size_t)EC*4;
  (void)ws_size; (void)n_in; (void)out_size;

  const int gN  = (N  + TPB - 1) / TPB;
  const int gEC = (EC + TPB - 1) / TPB;

  // ---- encoder ----
  k_zero8<<<1, 8, 0, stream>>>(red);
  k_enc1<<<gN, TPB, 0, stream>>>(x, enc_W1, enc_b1, encz, red, N);
  k_bn_fin<<<1, 4, 0, stream>>>(red, enc_g1, enc_be1, 1.f/(float)N);
  k_enc2<<<gN, TPB, 0, stream>>>(encz, enc_W2, enc_b2, red, N);
  k_bn_fin<<<1, 4, 0, stream>>>(red, enc_g2, enc_be2, 1.f/(float)N);
  k_enc3<<<gN, TPB, 0, stream>>>(encz, enc_W3, enc_b3, red, h0, N);

  // ---- GATv2 layers ----
  const int* eis[3] = { ei_p, ei_r, ei_c };
  const int  Es[3]  = { EP, EP, EC };
  for (int l = 0; l < 3; l++){
    const float* hin = (l == 0) ? h0 : (catx + (l-1)*12);
    int hstride = (l == 0) ? 12 : 36;
    k_node_lin<<<gN, TPB, 0, stream>>>(hin, hstride,
        conv_Wl + (size_t)l*144, conv_bl + (size_t)l*12,
        conv_Wr + (size_t)l*144, conv_br + (size_t)l*12, xl, xr, N);
    for (int r = 0; r < 3; r++){
      const int* ei = eis[r];
      int E = Es[r];
      int gE = (E + TPB - 1) / TPB;
      k_seg_init<<<gN, TPB, 0, stream>>>(smax, den, accb, r*4, N);
      k_edge1<<<gE, TPB, 0, stream>>>(ei, E, xl, xr, conv_att + (size_t)l*12 + r*4,
                                      etmp, smax, r*4);
      k_edge2<<<gE, TPB, 0, stream>>>(ei, E, etmp, smax, den);
      k_edge3<<<gE, TPB, 0, stream>>>(ei, E, xl, etmp, den, accb, r*4);
    }
    k_node_out<<<gN, TPB, 0, stream>>>(accb, conv_bias + (size_t)l*12, catx + l*12, N);
  }

  // ---- decoder ----
  k_zero8<<<1, 8, 0, stream>>>(red);
  int gW = (EC + 127) / 128;   // 128 rows per 256-thread block (16 rows/wave)
  k_dec1_wmma<<<gW, TPB, 0, stream>>>(ei_c, EC, catx, dec_W1, dec_b1, decz, red);
  k_bn_fin<<<1, 4, 0, stream>>>(red, dec_g1, dec_be1, 1.f/(float)EC);
  k_dec2<<<gEC, TPB, 0, stream>>>(decz, dec_W2, dec_b2, red, EC);
  k_bn_fin<<<1, 4, 0, stream>>>(red, dec_g2, dec_be2, 1.f/(float)EC);
  k_dec3<<<gEC, TPB, 0, stream>>>(decz, dec_W3, dec_b3, red, etmp, EC);
  int npairs = EC/2;
  k_softmax<<<(npairs + TPB - 1)/TPB, TPB, 0, stream>>>(etmp, (float*)d_out, npairs);
}